// Model_91216515432588
// MI455X (gfx1250) — compile-verified
//
#include <hip/hip_runtime.h>
#include <hip/hip_bf16.h>

// ---------------------------------------------------------------------------
// Sizes
#define Dn   12
#define G    1728            // Dn^3
#define NB   108             // G / 16 blocks for Cholesky
#define BSZ  4               // batch
#define NSYS 8               // B * 2 systems
#define HID  64

typedef __attribute__((ext_vector_type(2))) float v2f;
typedef __attribute__((ext_vector_type(8))) float v8f;

// ---------------------------------------------------------------------------
// Sparse row stencil of L = c0*I + DT + c5*DXX + c6*DYY  (h = 0.05)
__device__ __forceinline__ void sten_add(int* cols, float* vals, int& n, int c, float v) {
    for (int q = 0; q < n; ++q) if (cols[q] == c) { vals[q] += v; return; }
    cols[n] = c; vals[n] = v; ++n;
}

__device__ int row_stencil(int g, float c0, float c5, float c6, int* cols, float* vals) {
    const float invh = 20.0f, inv2h = 10.0f, invh2 = 400.0f;  // 1/h, 1/2h, 1/h^2
    int n = 0;
    int t = g / 144, r = (g / 12) % 12, k = g % 12;
    sten_add(cols, vals, n, g, c0);
    // DT = kron(d1, I144), stride 144
    if (t == 0)       { sten_add(cols,vals,n,g,-invh);        sten_add(cols,vals,n,g+144, invh); }
    else if (t == 11) { sten_add(cols,vals,n,g-144,-invh);    sten_add(cols,vals,n,g,     invh); }
    else              { sten_add(cols,vals,n,g-144,-inv2h);   sten_add(cols,vals,n,g+144, inv2h); }
    // c5 * DXX = kron(I, kron(d2, I12)), stride 12
    float s5 = c5 * invh2;
    if (r == 0)       { sten_add(cols,vals,n,g,s5);    sten_add(cols,vals,n,g+12,-2.f*s5); sten_add(cols,vals,n,g+24,s5); }
    else if (r == 11) { sten_add(cols,vals,n,g-24,s5); sten_add(cols,vals,n,g-12,-2.f*s5); sten_add(cols,vals,n,g,   s5); }
    else              { sten_add(cols,vals,n,g-12,s5); sten_add(cols,vals,n,g,   -2.f*s5); sten_add(cols,vals,n,g+12,s5); }
    // c6 * DYY = kron(I, kron(I, d2)), stride 1
    float s6 = c6 * invh2;
    if (k == 0)       { sten_add(cols,vals,n,g,s6);   sten_add(cols,vals,n,g+1,-2.f*s6); sten_add(cols,vals,n,g+2,s6); }
    else if (k == 11) { sten_add(cols,vals,n,g-2,s6); sten_add(cols,vals,n,g-1,-2.f*s6); sten_add(cols,vals,n,g,  s6); }
    else              { sten_add(cols,vals,n,g-1,s6); sten_add(cols,vals,n,g,  -2.f*s6); sten_add(cols,vals,n,g+1,s6); }
    return n;
}

__device__ __forceinline__ float elu(float x) { return x > 0.f ? x : (__expf(x) - 1.f); }

// ---------------------------------------------------------------------------
// Residual conv net (3x3 circular pad).  conv1: 12 -> 64 ch with ELU fused.
__global__ void conv_in_k(const float* __restrict__ x, const float* __restrict__ w,
                          const float* __restrict__ bias, float* __restrict__ out) {
    int idx = blockIdx.x * blockDim.x + threadIdx.x;
    if (idx >= BSZ * HID * 144) return;
    int hw = idx % 144, oc = (idx / 144) % HID, b = idx / (144 * HID);
    int h0 = hw / 12, w0 = hw % 12;
    float acc = bias[oc];
    for (int ic = 0; ic < 12; ++ic) {
        const float* xp = x + (b * 12 + ic) * 144;
        const float* wp = w + (oc * 12 + ic) * 9;
        for (int dh = 0; dh < 3; ++dh) {
            int hh = h0 + dh - 1; hh = (hh + 12) % 12;
            for (int dw = 0; dw < 3; ++dw) {
                int ww = w0 + dw - 1; ww = (ww + 12) % 12;
                acc += xp[hh * 12 + ww] * wp[dh * 3 + dw];
            }
        }
    }
    out[idx] = elu(acc);
}

// conv2: 64 -> 12 ch, + residual input
__global__ void conv_out_k(const float* __restrict__ hid, const float* __restrict__ w,
                           const float* __restrict__ bias, const float* __restrict__ resid,
                           float* __restrict__ out) {
    int idx = blockIdx.x * blockDim.x + threadIdx.x;
    if (idx >= BSZ * 12 * 144) return;
    int hw = idx % 144, oc = (idx / 144) % 12, b = idx / (144 * 12);
    int h0 = hw / 12, w0 = hw % 12;
    float acc = bias[oc];
    for (int ic = 0; ic < HID; ++ic) {
        const float* xp = hid + (b * HID + ic) * 144;
        const float* wp = w + (oc * HID + ic) * 9;
        for (int dh = 0; dh < 3; ++dh) {
            int hh = h0 + dh - 1; hh = (hh + 12) % 12;
            for (int dw = 0; dw < 3; ++dw) {
                int ww = w0 + dw - 1; ww = (ww + 12) % 12;
                acc += xp[hh * 12 + ww] * wp[dh * 3 + dw];
            }
        }
    }
    out[idx] = resid[idx] + acc;
}

// ---------------------------------------------------------------------------
// Batch-1 MLP layer: out[j] = elu(sum_k in[k]*W[k,j] + b[j])
__global__ void mlp_k(const float* __restrict__ in, const float* __restrict__ W,
                      const float* __restrict__ bias, float* __restrict__ out, int K, int N) {
    int j = blockIdx.x * blockDim.x + threadIdx.x;
    if (j >= N) return;
    float acc = bias[j];
    for (int k = 0; k < K; ++k) acc += in[k] * W[k * N + j];
    out[j] = elu(acc);
}

// Final layer: params = 2*tanh(h @ pw4 + pb4), also emit params + all-ones mask outputs.
__global__ void mlp_final_k(const float* __restrict__ in, const float* __restrict__ W,
                            const float* __restrict__ bias, float* __restrict__ params_ws,
                            float* __restrict__ out_params, float* __restrict__ out_mask) {
    int j = threadIdx.x;
    if (j >= 16) return;
    float acc = bias[j];
    for (int k = 0; k < 1024; ++k) acc += in[k] * W[k * 16 + j];
    float p = 2.0f * tanhf(acc);
    params_ws[j] = p;
    out_params[j] = p;
    out_mask[j] = 1.0f;
}

// ---------------------------------------------------------------------------
// Coefficients c0,c5,c6,rhs per (b,d,g) from params; also write uvp output.
__global__ void coeffs_k(const float* __restrict__ up, const float* __restrict__ up2,
                         const float* __restrict__ P, float* __restrict__ c0,
                         float* __restrict__ c5, float* __restrict__ c6,
                         float* __restrict__ rhs, float* __restrict__ out_uvp) {
    int idx = blockIdx.x * blockDim.x + threadIdx.x;
    if (idx >= BSZ * G) return;
    float uu = up[idx], vv = up2[idx];
    float A = uu * uu + vv * vv;
    int b = idx / G, g = idx % G;
    int o0 = (b * 2 + 0) * G + g, o1 = (b * 2 + 1) * G + g;
    // polys[p] = P[2p] + P[2p+1]*A ; dim0 -> polys 0..3, dim1 -> polys 4..7
    c0[o0]  = P[0]  + P[1]  * A;
    c5[o0]  = P[2]  + P[3]  * A;
    c6[o0]  = P[4]  + P[5]  * A;
    rhs[o0] = (P[6] + P[7]  * A) * vv;
    c0[o1]  = P[8]  + P[9]  * A;
    c5[o1]  = P[10] + P[11] * A;
    c6[o1]  = P[12] + P[13] * A;
    rhs[o1] = (P[14] + P[15] * A) * uu;
    out_uvp[o0] = uu;
    out_uvp[o1] = vv;
}

// ---------------------------------------------------------------------------
__global__ void zero_k(float* __restrict__ p, size_t n) {
    size_t i = (size_t)blockIdx.x * blockDim.x + threadIdx.x;
    size_t st = (size_t)gridDim.x * blockDim.x;
    for (; i < n; i += st) p[i] = 0.0f;
}

// Sparse normal-equation assembly: thread owns one row i of M[b,d]  (no atomics).
// M[i,j] = sum_g L[g,i] L[g,j]  (+1 on IV diagonal);  bvec[i] = sum_g L[g,i] rhs[g] (+iv).
__global__ void assemble_k(const float* __restrict__ c0a, const float* __restrict__ c5a,
                           const float* __restrict__ c6a, const float* __restrict__ rhsa,
                           const float* __restrict__ up, const float* __restrict__ up2,
                           float* __restrict__ M, float* __restrict__ bvec) {
    int row = blockIdx.x * blockDim.x + threadIdx.x;
    if (row >= NSYS * G) return;
    int i = row % G, bd = row / G, d = bd & 1, b = bd >> 1;
    float* Mrow = M + (size_t)bd * G * G + (size_t)i * G;
    const float* c0 = c0a + bd * G;
    const float* c5 = c5a + bd * G;
    const float* c6 = c6a + bd * G;
    int ti = i / 144, ri = (i / 12) % 12, ki = i % 12;
    // candidate rows g with L[g,i] possibly nonzero (one coord differs, <=2 away)
    int cand[13]; int nc = 0;
    cand[nc++] = i;
    for (int dt = -1; dt <= 1; dt += 2) { int t2 = ti + dt; if (t2 >= 0 && t2 < 12) cand[nc++] = i + dt * 144; }
    for (int dr = -2; dr <= 2; ++dr) { if (dr == 0) continue; int r2 = ri + dr; if (r2 >= 0 && r2 < 12) cand[nc++] = i + dr * 12; }
    for (int dk = -2; dk <= 2; ++dk) { if (dk == 0) continue; int k2 = ki + dk; if (k2 >= 0 && k2 < 12) cand[nc++] = i + dk; }
    float bacc = 0.0f;
    for (int q = 0; q < nc; ++q) {
        int g = cand[q];
        int cols[9]; float vals[9];
        int n = row_stencil(g, c0[g], c5[g], c6[g], cols, vals);
        float vi = 0.0f;
        for (int e = 0; e < n; ++e) if (cols[e] == i) vi = vals[e];
        if (vi != 0.0f) {
            for (int e = 0; e < n; ++e) Mrow[cols[e]] += vi * vals[e];
            bacc += vi * rhsa[bd * G + g];
        }
    }
    // IV constraint set: t==0 | r in {0,11} | k in {0,11}
    if (ti == 0 || ri == 0 || ri == 11 || ki == 0 || ki == 11) {
        Mrow[i] += 1.0f;
        bacc += (d == 0 ? up[b * G + i] : up2[b * G + i]);
    }
    bvec[bd * G + i] = bacc;
}

// ---------------------------------------------------------------------------
// Blocked Cholesky, 16-wide panels, NSYS independent systems in M[s] (G x G).
// (1) factor diagonal 16x16 block (one thread per system, L2-resident)
__global__ void chol_diag_k(float* __restrict__ M, int kb) {
    int s = threadIdx.x;
    if (s >= NSYS) return;
    float* A = M + (size_t)s * G * G;
    int base = kb * 16;
    for (int c = 0; c < 16; ++c) {
        float dv = sqrtf(A[(size_t)(base + c) * G + base + c]);
        A[(size_t)(base + c) * G + base + c] = dv;
        float inv = 1.0f / dv;
        for (int r = c + 1; r < 16; ++r) A[(size_t)(base + r) * G + base + c] *= inv;
        for (int cc = c + 1; cc < 16; ++cc) {
            float lcc = A[(size_t)(base + cc) * G + base + c];
            for (int r = cc; r < 16; ++r)
                A[(size_t)(base + r) * G + base + cc] -= A[(size_t)(base + r) * G + base + c] * lcc;
        }
    }
}

// (2) panel triangular solve:  X * Lk^T = B  for every row-block below diagonal.
//     16 lanes per block-task, each lane owns one row of the 16x16 tile.
__global__ void chol_panel_k(float* __restrict__ M, int kb) {
    int grp = (blockIdx.x * blockDim.x + threadIdx.x) >> 4;
    int lane = threadIdx.x & 15;
    int T = (NB - 1) - kb;
    if (grp >= NSYS * T) return;
    int s = grp / T;
    int bi = kb + 1 + (grp % T);
    float* A = M + (size_t)s * G * G;
    int rb = bi * 16 + lane, cb = kb * 16;
    float rowv[16];
#pragma unroll
    for (int c = 0; c < 16; ++c) rowv[c] = A[(size_t)rb * G + cb + c];
#pragma unroll
    for (int c = 0; c < 16; ++c) {
#pragma unroll
        for (int c2 = 0; c2 < 16; ++c2)
            if (c2 < c) rowv[c] -= rowv[c2] * A[(size_t)(cb + c) * G + cb + c2];
        rowv[c] /= A[(size_t)(cb + c) * G + cb + c];
    }
#pragma unroll
    for (int c = 0; c < 16; ++c) A[(size_t)rb * G + cb + c] = rowv[c];
}

// (3) trailing rank-16 update via f32 WMMA with 2x2 register blocking:
//     each wave owns a 32x32 macro-tile (4 sub-tiles), C(bi,bj) -= P(bi)*P(bj)^T.
//     16 x V_WMMA_F32_16X16X4_F32 per wave; the 4 WMMAs per k-slice are
//     independent (distinct accumulators) and reuse the same A/B operand pairs,
//     halving panel traffic from L2 vs one-tile-per-wave.
//     All boundary/triangle predicates are wave-uniform -> EXEC stays all-ones
//     for every WMMA (ISA requirement); row pointers are clamped in-range so
//     no OOB loads; invalid sub-tiles are never stored.
//     f32 WMMA has no A/B neg modifier -> negate A values at load (v_xor).
__global__ void chol_trail_k(float* __restrict__ M, int kb) {
    int wave = (blockIdx.x * blockDim.x + threadIdx.x) >> 5;
    int lane = threadIdx.x & 31;
    int T = (NB - 1) - kb;
    int T2 = (T + 1) >> 1;                       // macro-tiles per side
    int ntri = T2 * (T2 + 1) / 2;
    if (wave >= NSYS * ntri) return;             // wave-uniform
    int s = wave / ntri;
    int tri = wave % ntri;
    int J = 0, rem = tri;
    while (rem >= T2 - J) { rem -= T2 - J; ++J; }
    int I = J + rem;                             // I >= J : lower triangle of macro grid
    int bj0 = kb + 1 + 2 * J;
    int bi0 = kb + 1 + 2 * I;
    bool r1ok = (bi0 + 1 <= NB - 1);             // second row-block exists
    bool c1ok = (bj0 + 1 <= NB - 1);             // second col-block exists
    bool v01 = c1ok && (I > J);                  // (bi0, bj0+1): needs bi >= bj
    bool v10 = r1ok;                             // (bi0+1, bj0)
    bool v11 = r1ok && c1ok;                     // (bi0+1, bj0+1)
    int bi1 = r1ok ? bi0 + 1 : bi0;              // clamped (loads stay in range)
    int bj1 = c1ok ? bj0 + 1 : bj0;
    float* A = M + (size_t)s * G * G;
    int m = lane & 15, half = lane >> 4;
    size_t koff = (size_t)(kb * 16 + 2 * half);
    const float* rA0 = A + (size_t)(bi0 * 16 + m) * G + koff;
    const float* rA1 = A + (size_t)(bi1 * 16 + m) * G + koff;
    const float* rB0 = A + (size_t)(bj0 * 16 + m) * G + koff;
    const float* rB1 = A + (size_t)(bj1 * 16 + m) * G + koff;
    float* C00 = A + (size_t)(bi0 * 16 + 8 * half) * G + (size_t)(bj0 * 16 + m);
    float* C01 = A + (size_t)(bi0 * 16 + 8 * half) * G + (size_t)(bj1 * 16 + m);
    float* C10 = A + (size_t)(bi1 * 16 + 8 * half) * G + (size_t)(bj0 * 16 + m);
    float* C11 = A + (size_t)(bi1 * 16 + 8 * half) * G + (size_t)(bj1 * 16 + m);
    v8f a00 = {}, a01 = {}, a10 = {}, a11 = {};
#pragma unroll
    for (int r = 0; r < 8; ++r) a00[r] = C00[(size_t)r * G];
    if (v01) {
#pragma unroll
        for (int r = 0; r < 8; ++r) a01[r] = C01[(size_t)r * G];
    }
    if (v10) {
#pragma unroll
        for (int r = 0; r < 8; ++r) a10[r] = C10[(size_t)r * G];
    }
    if (v11) {
#pragma unroll
        for (int r = 0; r < 8; ++r) a11[r] = C11[(size_t)r * G];
    }
#pragma unroll
    for (int k0 = 0; k0 < 16; k0 += 4) {
        v2f pa0, pa1, pb0, pb1;
        pa0[0] = -rA0[k0]; pa0[1] = -rA0[k0 + 1];
        pa1[0] = -rA1[k0]; pa1[1] = -rA1[k0 + 1];
        pb0[0] =  rB0[k0]; pb0[1] =  rB0[k0 + 1];
        pb1[0] =  rB1[k0]; pb1[1] =  rB1[k0 + 1];
        a00 = __builtin_amdgcn_wmma_f32_16x16x4_f32(false, pa0, false, pb0, (short)0, a00, false, false);
        a01 = __builtin_amdgcn_wmma_f32_16x16x4_f32(false, pa0, false, pb1, (short)0, a01, false, false);
        a10 = __builtin_amdgcn_wmma_f32_16x16x4_f32(false, pa1, false, pb0, (short)0, a10, false, false);
        a11 = __builtin_amdgcn_wmma_f32_16x16x4_f32(false, pa1, false, pb1, (short)0, a11, false, false);
    }
#pragma unroll
    for (int r = 0; r < 8; ++r) C00[(size_t)r * G] = a00[r];
    if (v01) {
#pragma unroll
        for (int r = 0; r < 8; ++r) C01[(size_t)r * G] = a01[r];
    }
    if (v10) {
#pragma unroll
        for (int r = 0; r < 8; ++r) C10[(size_t)r * G] = a10[r];
    }
    if (v11) {
#pragma unroll
        for (int r = 0; r < 8; ++r) C11[(size_t)r * G] = a11[r];
    }
}

// ---------------------------------------------------------------------------
// Forward + backward substitution with L (lower triangle of M); one block/system.
__global__ void solve_k(const float* __restrict__ M, const float* __restrict__ bvecin,
                        float* __restrict__ xout, float* __restrict__ u0_out) {
    __shared__ float sb[G];
    int s = blockIdx.x, tid = threadIdx.x, nt = blockDim.x;
    const float* A = M + (size_t)s * G * G;
    for (int i = tid; i < G; i += nt) sb[i] = bvecin[s * G + i];
    __syncthreads();
    // L y = b
    for (int c = 0; c < G; ++c) {
        if (tid == 0) sb[c] = sb[c] / A[(size_t)c * G + c];
        __syncthreads();
        float xc = sb[c];
        for (int i = c + 1 + tid; i < G; i += nt) sb[i] -= A[(size_t)i * G + c] * xc;
        __syncthreads();
    }
    // L^T x = y
    for (int c = G - 1; c >= 0; --c) {
        if (tid == 0) sb[c] = sb[c] / A[(size_t)c * G + c];
        __syncthreads();
        float xc = sb[c];
        for (int i = tid; i < c; i += nt) sb[i] -= A[(size_t)c * G + i] * xc;
        __syncthreads();
    }
    for (int i = tid; i < G; i += nt) {
        float v = sb[i];
        xout[s * G + i] = v;
        u0_out[s * G + i] = v;
    }
}

// ---------------------------------------------------------------------------
// eps = L x - rhs (sparse row dot)
__global__ void eps_k(const float* __restrict__ c0a, const float* __restrict__ c5a,
                      const float* __restrict__ c6a, const float* __restrict__ rhsa,
                      const float* __restrict__ x, float* __restrict__ eps_out) {
    int row = blockIdx.x * blockDim.x + threadIdx.x;
    if (row >= NSYS * G) return;
    int g = row % G, bd = row / G;
    int cols[9]; float vals[9];
    int n = row_stencil(g, c0a[row], c5a[row], c6a[row], cols, vals);
    float acc = 0.0f;
    for (int e = 0; e < n; ++e) acc += vals[e] * x[bd * G + cols[e]];
    eps_out[row] = acc - rhsa[row];
}

// ---------------------------------------------------------------------------
extern "C" void kernel_launch(void* const* d_in, const int* in_sizes, int n_in,
                              void* d_out, int out_size, void* d_ws, size_t ws_size,
                              hipStream_t stream) {
    const float* u        = (const float*)d_in[0];
    const float* v        = (const float*)d_in[1];
    const float* r1w1     = (const float*)d_in[2];
    const float* r1b1     = (const float*)d_in[3];
    const float* r1w2     = (const float*)d_in[4];
    const float* r1b2     = (const float*)d_in[5];
    const float* r2w1     = (const float*)d_in[6];
    const float* r2b1     = (const float*)d_in[7];
    const float* r2w2     = (const float*)d_in[8];
    const float* r2b2     = (const float*)d_in[9];
    const float* param_in = (const float*)d_in[10];
    const float* pw1      = (const float*)d_in[11];
    const float* pb1      = (const float*)d_in[12];
    const float* pw2      = (const float*)d_in[13];
    const float* pb2      = (const float*)d_in[14];
    const float* pw3      = (const float*)d_in[15];
    const float* pb3      = (const float*)d_in[16];
    const float* pw4      = (const float*)d_in[17];
    const float* pb4      = (const float*)d_in[18];

    float* ws = (float*)d_ws;
    float* up     = ws;                 // 6912
    float* up2    = ws + 6912;          // 6912
    float* hid    = ws + 13824;         // 36864 (reused for both rnets)
    float* h1     = ws + 50688;         // 1024
    float* h2     = ws + 51712;         // 1024
    float* h3     = ws + 52736;         // 1024
    float* params = ws + 53760;         // 16
    float* c0     = ws + 53776;         // 13824
    float* c5     = ws + 67600;         // 13824
    float* c6     = ws + 81424;         // 13824
    float* rhs    = ws + 95248;         // 13824
    float* bvec   = ws + 109072;        // 13824 (x overwrites in place)
    float* M      = ws + 122896;        // 8 * 1728 * 1728 = 23,887,872 (~95.5 MB)

    float* out        = (float*)d_out;
    float* out_u0     = out;            // (B,2,D,D,D) = 13824
    float* out_uvp    = out + 13824;    // 13824
    float* out_eps    = out + 27648;    // 13824
    float* out_params = out + 41472;    // 16
    float* out_mask   = out + 41488;    // 16

    // residual conv nets
    conv_in_k <<<(BSZ * HID * 144 + 255) / 256, 256, 0, stream>>>(u, r1w1, r1b1, hid);
    conv_out_k<<<(BSZ * 12  * 144 + 255) / 256, 256, 0, stream>>>(hid, r1w2, r1b2, u, up);
    conv_in_k <<<(BSZ * HID * 144 + 255) / 256, 256, 0, stream>>>(v, r2w1, r2b1, hid);
    conv_out_k<<<(BSZ * 12  * 144 + 255) / 256, 256, 0, stream>>>(hid, r2w2, r2b2, v, up2);

    // param net
    mlp_k<<<4, 256, 0, stream>>>(param_in, pw1, pb1, h1, 128, 1024);
    mlp_k<<<4, 256, 0, stream>>>(h1, pw2, pb2, h2, 1024, 1024);
    mlp_k<<<4, 256, 0, stream>>>(h2, pw3, pb3, h3, 1024, 1024);
    mlp_final_k<<<1, 32, 0, stream>>>(h3, pw4, pb4, params, out_params, out_mask);

    // coefficients + uvp output
    coeffs_k<<<(BSZ * G + 255) / 256, 256, 0, stream>>>(up, up2, params, c0, c5, c6, rhs, out_uvp);

    // sparse normal-equation assembly
    size_t Mtot = (size_t)NSYS * G * G;
    zero_k<<<4096, 256, 0, stream>>>(M, Mtot);
    assemble_k<<<(NSYS * G + 255) / 256, 256, 0, stream>>>(c0, c5, c6, rhs, up, up2, M, bvec);

    // blocked Cholesky (WMMA trailing updates, 2x2 macro-tiles per wave)
    for (int kb = 0; kb < NB; ++kb) {
        chol_diag_k<<<1, 32, 0, stream>>>(M, kb);
        int T = (NB - 1) - kb;
        if (T > 0) {
            int groups = NSYS * T;                       // 16-lane tasks
            chol_panel_k<<<(groups * 16 + 255) / 256, 256, 0, stream>>>(M, kb);
            int T2 = (T + 1) >> 1;
            long ntri  = (long)T2 * (T2 + 1) / 2;
            long waves = (long)NSYS * ntri;              // one wave per 32x32 macro-tile
            unsigned blocks = (unsigned)((waves * 32 + 255) / 256);
            chol_trail_k<<<blocks, 256, 0, stream>>>(M, kb);
        }
    }

    // triangular solves -> x (in bvec) and u0 output
    solve_k<<<NSYS, 256, 0, stream>>>(M, bvec, bvec, out_u0);

    // residual eps = L x - rhs
    eps_k<<<(NSYS * G + 255) / 256, 256, 0, stream>>>(c0, c5, c6, rhs, bvec, out_eps);
}